// GraphAttentionLayer_54417235640595
// MI455X (gfx1250) — compile-verified
//
#include <hip/hip_runtime.h>
#include <hip/hip_bf16.h>

#define B_    8
#define N_    2048
#define FIN   256
#define FOUT  128
#define ALPHA 0.2f
#define NEGV  -9000000000000000.0f

typedef __attribute__((ext_vector_type(2))) float v2f;
typedef __attribute__((ext_vector_type(8))) float v8f;

// ---------------------------------------------------------------------------
// Kernel 1: h = x @ W   (M = B*N = 16384, K = 256, N = 128)
// One wave computes a 16x128 output tile with 8 v8f accumulators, using
// V_WMMA_F32_16X16X4_F32.  A-frag: lane l holds rows m=l&15, K-pair (l>>4)*2.
// ---------------------------------------------------------------------------
__global__ __launch_bounds__(256) void gat_gemm_h(const float* __restrict__ x,
                                                  const float* __restrict__ W,
                                                  float* __restrict__ h) {
    const int lane = threadIdx.x & 31;
    const int wave = threadIdx.x >> 5;
    const int r0   = (blockIdx.x * 8 + wave) * 16;   // first row of tile
    const int m    = lane & 15;
    const int kg   = (lane >> 4) << 1;               // 0 or 2

    v8f acc[8] = {};

    const float* xrow = x + (size_t)(r0 + m) * FIN;

    for (int k0 = 0; k0 < FIN; k0 += 4) {
        v2f av;
        av.x = xrow[k0 + kg];
        av.y = xrow[k0 + kg + 1];
        const float* wr0 = W + (size_t)(k0 + kg) * FOUT;
        const float* wr1 = wr0 + FOUT;
#pragma unroll
        for (int ct = 0; ct < 8; ++ct) {
            v2f bv;
            bv.x = wr0[ct * 16 + m];
            bv.y = wr1[ct * 16 + m];
            acc[ct] = __builtin_amdgcn_wmma_f32_16x16x4_f32(
                false, av, false, bv, (short)0, acc[ct], false, false);
        }
    }

    // D layout: lane -> col n = lane&15, row = v + 8*(lane>>4)
    const int rowoff = (lane >> 4) * 8;
#pragma unroll
    for (int ct = 0; ct < 8; ++ct)
#pragma unroll
        for (int v = 0; v < 8; ++v)
            h[(size_t)(r0 + rowoff + v) * FOUT + ct * 16 + m] = acc[ct][v];
}

// ---------------------------------------------------------------------------
// Kernel 2: f[r] = h[r,:] . a[:128] ;  g[r] = h[r,:] . a[128:256]
// ---------------------------------------------------------------------------
__global__ __launch_bounds__(128) void gat_fg(const float* __restrict__ h,
                                              const float* __restrict__ a,
                                              float* __restrict__ f,
                                              float* __restrict__ g) {
    const int r = blockIdx.x * blockDim.x + threadIdx.x;   // 0 .. 16383
    const float* hr = h + (size_t)r * FOUT;
    float sf = 0.f, sg = 0.f;
#pragma unroll 8
    for (int n = 0; n < FOUT; ++n) {
        const float hv = hr[n];
        sf += hv * a[n];
        sg += hv * a[FOUT + n];
    }
    f[r] = sf;
    g[r] = sg;
}

// ---------------------------------------------------------------------------
// Kernel 3 (fused): masked-softmax attention + (attn @ h) + ELU.
// One block per (b, 16-row tile); 4 waves each own a 512-wide j quarter.
// adj stripe (16 x 2048 i32 = 128 KB) is pulled from HBM once; the max-pass /
// sum-pass / WMMA-pass re-reads hit WGP$/L2.
// ---------------------------------------------------------------------------
__global__ __launch_bounds__(128) void gat_attn_fused(const int* __restrict__ adj,
                                                      const float* __restrict__ f,
                                                      const float* __restrict__ g,
                                                      const float* __restrict__ h,
                                                      float* __restrict__ out) {
    __shared__ float gsh[N_];              // 8 KB : g[b, :]
    __shared__ float fsh[16];              // f tile
    __shared__ float wm[4][16];            // per-wave row max
    __shared__ float ws[4][16];            // per-wave row sum
    __shared__ float Mrow[16];             // combined row max
    __shared__ float Sinv[16];             // combined 1/row-sum
    __shared__ float part[4][16 * FOUT];   // 32 KB wave partials

    const int tid  = threadIdx.x;
    const int lane = tid & 31;
    const int wave = tid >> 5;             // 0..3
    const int b    = blockIdx.x >> 7;
    const int i0   = (blockIdx.x & 127) * 16;

    // ---- stage g row and f tile into LDS --------------------------------
    const float* gb = g + (size_t)b * N_;
    for (int j = tid; j < N_; j += 128) gsh[j] = gb[j];
    if (tid < 16) fsh[tid] = f[b * N_ + i0 + tid];
    __syncthreads();

    const int jq = wave * (N_ / 4);        // this wave's quarter start
    const size_t adj_base = (size_t)(b * N_ + i0) * N_;

    // ---- phase 1a: per-row max over quarter (coalesced, no exp) ---------
    float mloc[16];
#pragma unroll 1
    for (int r = 0; r < 16; ++r) {
        const int* arow = adj + adj_base + (size_t)r * N_;
        const float fi = fsh[r];
        float mr = -3.4e38f;
        for (int j = jq + lane; j < jq + (N_ / 4); j += 32) {
            __builtin_prefetch(arow + j + 256, 0, 1);
            const float z = fi + gsh[j];
            const float e = (arow[j] > 0) ? (z >= 0.f ? z : ALPHA * z) : NEGV;
            mr = fmaxf(mr, e);
        }
        mloc[r] = mr;
    }
#pragma unroll
    for (int r = 0; r < 16; ++r)
#pragma unroll
        for (int off = 16; off > 0; off >>= 1)
            mloc[r] = fmaxf(mloc[r], __shfl_xor(mloc[r], off, 32));

    // ---- phase 1b: per-row sum of exp(e - mr) over quarter --------------
#pragma unroll 1
    for (int r = 0; r < 16; ++r) {
        const int* arow = adj + adj_base + (size_t)r * N_;
        const float fi = fsh[r];
        const float mr = mloc[r];
        float s = 0.f;
        for (int j = jq + lane; j < jq + (N_ / 4); j += 32) {
            const float z = fi + gsh[j];
            const float e = (arow[j] > 0) ? (z >= 0.f ? z : ALPHA * z) : NEGV;
            s += __expf(e - mr);
        }
#pragma unroll
        for (int off = 16; off > 0; off >>= 1) s += __shfl_xor(s, off, 32);
        if (lane == 0) { wm[wave][r] = mr; ws[wave][r] = s; }
    }
    __syncthreads();

    // ---- combine quarter stats across the 4 waves -----------------------
    if (tid < 16) {
        const float m0 = wm[0][tid], m1 = wm[1][tid], m2 = wm[2][tid], m3 = wm[3][tid];
        const float M  = fmaxf(fmaxf(m0, m1), fmaxf(m2, m3));
        const float S  = ws[0][tid] * __expf(m0 - M) + ws[1][tid] * __expf(m1 - M) +
                         ws[2][tid] * __expf(m2 - M) + ws[3][tid] * __expf(m3 - M);
        Mrow[tid] = M;
        Sinv[tid] = 1.0f / S;
    }
    __syncthreads();

    // ---- phase 2: WMMA  acc += exp(e - M) * h  over quarter -------------
    const int m  = lane & 15;
    const int kg = (lane >> 4) << 1;       // 0 or 2
    const float fi = fsh[m];
    const float Mi = Mrow[m];
    const int*   arow = adj + adj_base + (size_t)m * N_;
    const float* hb   = h + (size_t)b * N_ * FOUT;

    v8f acc[8] = {};

    for (int j0 = jq; j0 < jq + (N_ / 4); j0 += 4) {
        const int j = j0 + kg;
        const float z0 = fi + gsh[j];
        const float e0 = (arow[j] > 0) ? (z0 >= 0.f ? z0 : ALPHA * z0) : NEGV;
        const float z1 = fi + gsh[j + 1];
        const float e1 = (arow[j + 1] > 0) ? (z1 >= 0.f ? z1 : ALPHA * z1) : NEGV;

        v2f av;
        av.x = __expf(e0 - Mi);            // 1/S applied in epilogue
        av.y = __expf(e1 - Mi);

        const float* h0 = hb + (size_t)j * FOUT;
        const float* h1 = h0 + FOUT;
#pragma unroll
        for (int ct = 0; ct < 8; ++ct) {
            v2f bv;
            bv.x = h0[ct * 16 + m];
            bv.y = h1[ct * 16 + m];
            acc[ct] = __builtin_amdgcn_wmma_f32_16x16x4_f32(
                false, av, false, bv, (short)0, acc[ct], false, false);
        }
    }

    // ---- dump wave partial (D layout) to LDS ----------------------------
    const int rowoff = (lane >> 4) * 8;
    float* my = &part[wave][0];
#pragma unroll
    for (int ct = 0; ct < 8; ++ct)
#pragma unroll
        for (int v = 0; v < 8; ++v)
            my[(rowoff + v) * FOUT + ct * 16 + m] = acc[ct][v];
    __syncthreads();

    // ---- reduce partials, normalize by 1/S, ELU, store ------------------
    for (int idx = tid; idx < 16 * FOUT; idx += 128) {
        const int row = idx >> 7;
        const int col = idx & 127;
        float sum = part[0][idx] + part[1][idx] + part[2][idx] + part[3][idx];
        sum *= Sinv[row];
        const float o = sum > 0.f ? sum : (__expf(sum) - 1.0f);
        out[(size_t)(b * N_ + i0 + row) * FOUT + col] = o;
    }
}

// ---------------------------------------------------------------------------
extern "C" void kernel_launch(void* const* d_in, const int* in_sizes, int n_in,
                              void* d_out, int out_size, void* d_ws, size_t ws_size,
                              hipStream_t stream) {
    (void)in_sizes; (void)n_in; (void)out_size; (void)ws_size;

    const float* x   = (const float*)d_in[0];   // (8, 2048, 256) f32
    const int*   adj = (const int*)  d_in[1];   // (8, 2048, 2048) i32
    const float* W   = (const float*)d_in[2];   // (256, 128) f32
    const float* a   = (const float*)d_in[3];   // (256, 1) f32
    float*       out = (float*)d_out;           // (8, 2048, 128) f32

    char* ws = (char*)d_ws;
    float* h = (float*)ws;                               // 16384*128 f32 = 8 MB
    float* f = (float*)(ws + (size_t)16384 * FOUT * 4);  // 16384 f32
    float* g = f + 16384;

    // 1) h = x @ W  (128 blocks x 8 waves, one 16x128 tile per wave)
    gat_gemm_h<<<128, 256, 0, stream>>>(x, W, h);
    // 2) f, g projections
    gat_fg<<<128, 128, 0, stream>>>(h, a, f, g);
    // 3) fused masked softmax + attention aggregation + ELU (1024 tiles)
    gat_attn_fused<<<1024, 128, 0, stream>>>(adj, f, g, h, out);
}